// BaseMultiLora_34007551050174
// MI455X (gfx1250) — compile-verified
//
#include <hip/hip_runtime.h>

// Multi-LoRA base forward:  y[b] = x[b] @ w[adapter_ids[b]].T
// x: [8, 2048, 4096] f32   w: [16, 4096, 4096] f32   y: [8, 2048, 4096] f32
//
// Memory-bound (~1 GB min traffic vs 0.55 TFLOP -> ~45us floor at 23.3 TB/s).
// fp32 WMMA 16x16x4, 128x128 block tiles, register double-buffered K chunks:
// next chunk's global_load_b128s stay in flight behind the 64-WMMA compute
// phase; LDS fragment reads are bank-conflict-free ds_load_b64.

typedef float v2f __attribute__((ext_vector_type(2)));
typedef float v4f __attribute__((ext_vector_type(4)));
typedef float v8f __attribute__((ext_vector_type(8)));

namespace {
constexpr int kB    = 8;
constexpr int kS    = 2048;
constexpr int kDin  = 4096;
constexpr int kDout = 4096;

constexpr int BM  = 128;      // block tile M (rows of x)
constexpr int BN  = 128;      // block tile N (rows of w = cols of y)
constexpr int KC  = 32;       // K chunk staged in LDS
constexpr int LDA = KC + 4;   // 36 floats: 16B-aligned float4 rows + bank spread
}

__global__ __launch_bounds__(256, 2)
void multilora_wmma_f32(const float* __restrict__ x,
                        const float* __restrict__ w,
                        const int*   __restrict__ adapter_ids,
                        float*       __restrict__ y)
{
    __shared__ float As[BM * LDA];   // x tile [128 rows][36]
    __shared__ float Bs[BN * LDA];   // w tile [128 rows][36]

    const int b   = blockIdx.z;
    const int m0  = blockIdx.y * BM;
    const int n0  = blockIdx.x * BN;
    const int aid = adapter_ids[b];

    const float* __restrict__ xb = x + (size_t)b   * kS    * kDin;
    const float* __restrict__ wb = w + (size_t)aid * kDout * kDin;
    float*       __restrict__ yb = y + (size_t)b   * kS    * kDout;

    const int tid  = threadIdx.x;   // 0..255
    const int lane = tid & 31;
    const int wave = tid >> 5;      // 0..7
    const int wm   = wave & 1;      // 2 waves along M, 64 rows each
    const int wn   = wave >> 1;     // 4 waves along N, 32 cols each
    const int lo   = lane & 15;
    const int hi   = lane >> 4;

    // Per-thread staging coordinates: 8 lanes cover one row
    // (8 x float4 = 128B contiguous) -> fully coalesced b128 loads.
    int srow[4], skq[4];
    #pragma unroll
    for (int i = 0; i < 4; ++i) {
        const int idx = tid + 256 * i;   // 0..1023
        srow[i] = idx >> 3;              // 0..127
        skq[i]  = (idx & 7) << 2;        // 0,4,...,28
    }

    v8f acc[4][2] = {};     // 4 M-subtiles x 2 N-subtiles of 16x16 f32

    // Preload chunk 0 into registers.
    v4f ra[4], rb[4];
    #pragma unroll
    for (int i = 0; i < 4; ++i) {
        ra[i] = *(const v4f*)(xb + (size_t)(m0 + srow[i]) * kDin + skq[i]);
        rb[i] = *(const v4f*)(wb + (size_t)(n0 + srow[i]) * kDin + skq[i]);
    }

    for (int k0 = 0; k0 < kDin; k0 += KC) {
        __syncthreads();    // all waves done reading LDS from previous chunk

        // Drain registers -> LDS (implicit s_wait_loadcnt lands here).
        #pragma unroll
        for (int i = 0; i < 4; ++i) {
            *(v4f*)&As[srow[i] * LDA + skq[i]] = ra[i];
            *(v4f*)&Bs[srow[i] * LDA + skq[i]] = rb[i];
        }
        __syncthreads();

        // Kick off chunk k+1 loads; they fly behind the 64 WMMAs below.
        const int k1 = k0 + KC;
        if (k1 < kDin) {
            #pragma unroll
            for (int i = 0; i < 4; ++i) {
                ra[i] = *(const v4f*)(xb + (size_t)(m0 + srow[i]) * kDin + k1 + skq[i]);
                rb[i] = *(const v4f*)(wb + (size_t)(n0 + srow[i]) * kDin + k1 + skq[i]);
            }
        }
        // Warm L2 two chunks ahead (global_prefetch_b8).
        if (k0 + 2 * KC < kDin) {
            const int prow = tid >> 1;
            const int phalf = (tid & 1) << 4;
            __builtin_prefetch(xb + (size_t)(m0 + prow) * kDin + k0 + 2 * KC + phalf, 0, 3);
            __builtin_prefetch(wb + (size_t)(n0 + prow) * kDin + k0 + 2 * KC + phalf, 0, 3);
        }

        // 8 WMMA K-steps of 4 per chunk.
        #pragma unroll
        for (int kk = 0; kk < KC; kk += 4) {
            const int kc = kk + 2 * hi;        // per-lane K offset (A/B layout)

            v2f af[4];
            #pragma unroll
            for (int i = 0; i < 4; ++i)
                af[i] = *(const v2f*)&As[(wm * 64 + i * 16 + lo) * LDA + kc];

            v2f bf[2];
            #pragma unroll
            for (int j = 0; j < 2; ++j)
                bf[j] = *(const v2f*)&Bs[(wn * 32 + j * 16 + lo) * LDA + kc];

            #pragma unroll
            for (int i = 0; i < 4; ++i)
                #pragma unroll
                for (int j = 0; j < 2; ++j)
                    acc[i][j] = __builtin_amdgcn_wmma_f32_16x16x4_f32(
                        /*neg_a=*/false, af[i],
                        /*neg_b=*/false, bf[j],
                        /*c_mod=*/(short)0, acc[i][j],
                        /*reuse_a=*/false, /*reuse_b=*/false);
        }
    }

    // D layout: VGPR v holds row (v + 8*hi) of the 16x16 tile, lane&15 = col.
    #pragma unroll
    for (int i = 0; i < 4; ++i) {
        const int mbase = m0 + wm * 64 + i * 16 + 8 * hi;
        #pragma unroll
        for (int j = 0; j < 2; ++j) {
            const int ncol = n0 + wn * 32 + j * 16 + lo;
            #pragma unroll
            for (int v = 0; v < 8; ++v)
                yb[(size_t)(mbase + v) * kDout + ncol] = acc[i][j][v];
        }
    }
}

extern "C" void kernel_launch(void* const* d_in, const int* in_sizes, int n_in,
                              void* d_out, int out_size, void* d_ws, size_t ws_size,
                              hipStream_t stream) {
    const float* x   = (const float*)d_in[0];
    const float* w   = (const float*)d_in[1];
    const int*   ids = (const int*)d_in[2];
    float*       y   = (float*)d_out;

    dim3 grid(kDout / BN, kS / BM, kB);   // (32, 16, 8)
    multilora_wmma_f32<<<grid, 256, 0, stream>>>(x, w, ids, y);

    (void)in_sizes; (void)n_in; (void)out_size; (void)d_ws; (void)ws_size;
}